// ChamferDistance_43748536877744
// MI455X (gfx1250) — compile-verified
//
#include <hip/hip_runtime.h>
#include <math.h>

typedef float v2f __attribute__((ext_vector_type(2)));
typedef float v8f __attribute__((ext_vector_type(8)));

#define CB 4
#define CN 4096
#define TILES (CN / 16)                      // 256 column tiles per batch
#define NGRP  (TILES / 4)                    // 64 groups of 4 tiles
#define BLOCKS_PER_PASS (CB * (TILES / 8))   // 8 waves/block -> 128 blocks/pass

struct Tile { float x, y, z; };

__device__ __forceinline__ void load_group(const float* __restrict__ cbase,
                                           int grp, int lrow, Tile t[4]) {
#pragma unroll
    for (int k = 0; k < 4; ++k) {
        const int col = (grp * 4 + k) * 16 + lrow;
        t[k].x = cbase[col * 3 + 0];
        t[k].y = cbase[col * 3 + 1];
        t[k].z = cbase[col * 3 + 2];
    }
}

__device__ __forceinline__ void process_group(const Tile t[4], v2f a, v8f c,
                                              bool lo, float rmin[8]) {
#pragma unroll
    for (int k = 0; k < 4; ++k) {
        const float pn = t[k].x * t[k].x + t[k].y * t[k].y + t[k].z * t[k].z;
        v2f bm;
        bm.x = lo ? (-2.0f * t[k].x) : (-2.0f * t[k].z);  // K=0 | K=2
        bm.y = lo ? (-2.0f * t[k].y) : pn;                // K=1 | K=3
        const v8f d = __builtin_amdgcn_wmma_f32_16x16x4_f32(
            false, a, false, bm, (short)0, c, false, false);
#pragma unroll
        for (int r = 0; r < 8; ++r) rmin[r] = fminf(rmin[r], d[r]);
    }
}

// lane_src = ((lane & 0x1F) | 0) ^ mask  ->  group-of-32 xor swizzle
__device__ __forceinline__ float xor_min(float v, float w) { return fminf(v, w); }

// ---------------------------------------------------------------------------
// One fused kernel, two logical passes (roles of preds/targets swapped):
//   dist^2 tile = A(16x4:[r,1]) x B(4x16:[-2c, ||c||^2]) + C(||r||^2)
// Each wave owns 16 query rows; ping-pong double-buffered loop over all
// 256 column tiles (4 tiles/group); row-min in 8 VGPRs; no atomics, no LDS.
// ---------------------------------------------------------------------------
__global__ __launch_bounds__(256) void chamfer_min_kernel(
    const float* __restrict__ preds,    // (B, N, 3)
    const float* __restrict__ targets,  // (B, N, 3)
    float* __restrict__ mins)           // (2, B, N)
{
    const int  lane   = threadIdx.x & 31;
    const int  wave   = threadIdx.x >> 5;
    const bool second = blockIdx.x >= BLOCKS_PER_PASS;
    const int  bid    = second ? (int)blockIdx.x - BLOCKS_PER_PASS : (int)blockIdx.x;

    const float* rows = second ? preds   : targets;
    const float* cols = second ? targets : preds;
    float* outmin     = mins + (second ? (size_t)CB * CN : 0);

    const int blocksPerBatch = TILES / 8;  // 32
    const int b       = bid / blocksPerBatch;
    const int rowTile = (bid % blocksPerBatch) * 8 + wave;
    const int lrow    = lane & 15;
    const int row     = rowTile * 16 + lrow;
    const bool lo     = lane < 16;

    const float* rbase = rows + (size_t)b * CN * 3;
    const float* cbase = cols + (size_t)b * CN * 3;

    // ---- A fragment: [rx, ry, rz, 1] (fixed) ----
    const float tx = rbase[row * 3 + 0];
    const float ty = rbase[row * 3 + 1];
    const float tz = rbase[row * 3 + 2];
    const float tn = tx * tx + ty * ty + tz * tz;
    v2f a;
    a.x = lo ? tx : tz;    // K=0 | K=2
    a.y = lo ? ty : 1.0f;  // K=1 | K=3

    // ---- C fragment: C[m][n] = ||r_m||^2, m = r + 8*(lane>=16) (fixed) ----
    v8f c;
    const int srcbase = (lane & 16) ? 8 : 0;
#pragma unroll
    for (int r = 0; r < 8; ++r) c[r] = __shfl(tn, srcbase + r, 32);

    float rmin[8];
#pragma unroll
    for (int r = 0; r < 8; ++r) rmin[r] = __builtin_inff();

    // ---- ping-pong double-buffered tile-group loop ----
    Tile ga[4], gb[4];
    load_group(cbase, 0, lrow, ga);
    load_group(cbase, 1, lrow, gb);
    for (int grp = 0; grp < NGRP; grp += 2) {
        process_group(ga, a, c, lo, rmin);
        load_group(cbase, (grp + 2) & (NGRP - 1), lrow, ga);  // wrap: valid, unused
        process_group(gb, a, c, lo, rmin);
        load_group(cbase, (grp + 3) & (NGRP - 1), lrow, gb);
    }

    // ---- row-min across each 16-lane half via ds_swizzle xor steps ----
#pragma unroll
    for (int r = 0; r < 8; ++r) {
        float v = rmin[r];
        v = xor_min(v, __uint_as_float(__builtin_amdgcn_ds_swizzle(__float_as_uint(v), 0x041F))); // xor 1
        v = xor_min(v, __uint_as_float(__builtin_amdgcn_ds_swizzle(__float_as_uint(v), 0x081F))); // xor 2
        v = xor_min(v, __uint_as_float(__builtin_amdgcn_ds_swizzle(__float_as_uint(v), 0x101F))); // xor 4
        v = xor_min(v, __uint_as_float(__builtin_amdgcn_ds_swizzle(__float_as_uint(v), 0x201F))); // xor 8
        rmin[r] = fmaxf(v, 0.0f);  // clamp fp-expansion negatives once
    }
    if (lrow == 0) {
        const int half8 = (lane >> 4) * 8;  // lane 0: rows 0..7, lane 16: rows 8..15
        float4* dst = (float4*)&outmin[(size_t)b * CN + rowTile * 16 + half8];
        dst[0] = make_float4(rmin[0], rmin[1], rmin[2], rmin[3]);
        dst[1] = make_float4(rmin[4], rmin[5], rmin[6], rmin[7]);
    }
}

// ---------------------------------------------------------------------------
// Finalize: out = (sum over both passes of sqrt(min)) / (B*N)
// ---------------------------------------------------------------------------
__global__ __launch_bounds__(1024) void chamfer_finalize_kernel(
    const float* __restrict__ mins, float* __restrict__ out)
{
    __shared__ float sdata[1024];
    float s = 0.0f;
    for (int i = threadIdx.x; i < 2 * CB * CN; i += 1024)
        s += sqrtf(mins[i]);
    sdata[threadIdx.x] = s;
    __syncthreads();
    for (int stride = 512; stride > 0; stride >>= 1) {
        if ((int)threadIdx.x < stride)
            sdata[threadIdx.x] += sdata[threadIdx.x + stride];
        __syncthreads();
    }
    if (threadIdx.x == 0) out[0] = sdata[0] / (float)(CB * CN);
}

// ---------------------------------------------------------------------------
extern "C" void kernel_launch(void* const* d_in, const int* in_sizes, int n_in,
                              void* d_out, int out_size, void* d_ws, size_t ws_size,
                              hipStream_t stream) {
    const float* preds   = (const float*)d_in[0];
    const float* targets = (const float*)d_in[1];
    float* out  = (float*)d_out;
    float* mins = (float*)d_ws;  // 2*B*N floats = 128 KB scratch

    chamfer_min_kernel<<<2 * BLOCKS_PER_PASS, 256, 0, stream>>>(preds, targets, mins);
    chamfer_finalize_kernel<<<1, 1024, 0, stream>>>(mins, out);
}